// ChunkwiseRetention_73538430042347
// MI455X (gfx1250) — compile-verified
//
#include <hip/hip_runtime.h>
#include <hip/hip_bf16.h>

typedef __attribute__((ext_vector_type(2))) float v2f;
typedef __attribute__((ext_vector_type(8))) float v8f;
typedef __attribute__((ext_vector_type(4))) unsigned int u32x4;
typedef __attribute__((ext_vector_type(8))) int i32x8;
typedef __attribute__((ext_vector_type(4))) int i32x4;

#define B_ 8
#define S_ 1024
#define D_ 512
#define MD_ ((size_t)B_ * S_ * D_)   // 4194304 elements per matrix
#define GAMMA_F 0.9865f

// ---- gfx1250 async / TDM feature detection --------------------------------
#if defined(__gfx1250__) && __has_builtin(__builtin_amdgcn_global_load_async_to_lds_b128) && \
    __has_builtin(__builtin_amdgcn_s_wait_asynccnt)
#define USE_ASYNC_LDS 1
#else
#define USE_ASYNC_LDS 0
#endif

#if defined(__gfx1250__) && __has_builtin(__builtin_amdgcn_tensor_load_to_lds) && \
    __has_builtin(__builtin_amdgcn_s_wait_tensorcnt)
#define USE_TDM 1
#else
#define USE_TDM 0
#endif

#if USE_ASYNC_LDS || USE_TDM
// addrspace-qualified vector pointee types per the builtin's real signature:
// param0: i32x4 addrspace(1)* (global src), param1: i32x4 addrspace(3)* (LDS dst)
typedef __attribute__((address_space(1))) i32x4 gas_v4i;
typedef __attribute__((address_space(3))) i32x4 las_v4i;
__device__ __forceinline__ unsigned lds_offset(const void* p) {
    return (unsigned)(size_t)(__attribute__((address_space(3))) const char*)p;
}
#endif

__device__ __forceinline__ void async_copy16(const float* src, float* dst_lds) {
#if USE_ASYNC_LDS
    __builtin_amdgcn_global_load_async_to_lds_b128(
        (gas_v4i*)(float*)src, (las_v4i*)dst_lds, 0, 0);
#else
    *(float4*)dst_lds = *(const float4*)src;
#endif
}

__device__ __forceinline__ void async_wait() {
#if USE_ASYNC_LDS
    __builtin_amdgcn_s_wait_asynccnt(0);
#endif
}

__device__ __forceinline__ v8f wmma4(v2f a, v2f b, v8f c) {
    // D = A(16x4 f32) * B(4x16 f32) + C(16x16 f32)
    return __builtin_amdgcn_wmma_f32_16x16x4_f32(
        false, a, false, b, (short)0, c, false, false);
}

// ---------------------------------------------------------------------------
// Kernel 1: projections Q=xq@Wq, K=xk@Wk, V=xv@Wv
// grid (M/64=128, N/64=8, 3), block 128 (4 waves); each wave -> 32x32 tile.
// ---------------------------------------------------------------------------
__global__ void __launch_bounds__(128)
proj_gemm_kernel(const float* __restrict__ xq, const float* __restrict__ xk,
                 const float* __restrict__ xv, const float* __restrict__ wq,
                 const float* __restrict__ wk, const float* __restrict__ wv,
                 float* __restrict__ ws) {
    __shared__ float As[64 * 32];   // [row][k]
    __shared__ float Bs[32 * 64];   // [k][col]

    const int mt = blockIdx.x, nt = blockIdx.y, which = blockIdx.z;
    const float* X = (which == 0) ? xq : (which == 1) ? xk : xv;
    const float* W = (which == 0) ? wq : (which == 1) ? wk : wv;
    float* O = ws + (size_t)which * MD_;

    const int tid  = threadIdx.x;
    const int lane = tid & 31;
    const int wave = tid >> 5;       // 0..3
    const int half = lane >> 4;      // 0/1
    const int l15  = lane & 15;
    const int wr = (wave & 1) * 32;  // wave row offset in 64x64 tile
    const int wc = (wave >> 1) * 32; // wave col offset

    const v8f vzero = {0.f,0.f,0.f,0.f,0.f,0.f,0.f,0.f};
    v8f acc[2][2] = {{vzero, vzero}, {vzero, vzero}};

    const int row0 = mt * 64, col0 = nt * 64;

    for (int kb = 0; kb < D_; kb += 32) {
        // stage A: 64 rows x 32 k  (512 x 16B, 4 per thread) via async LDS DMA
        for (int i = tid; i < 64 * 8; i += 128) {
            int r = i >> 3, c4 = i & 7;
            async_copy16(X + (size_t)(row0 + r) * D_ + kb + c4 * 4,
                         As + r * 32 + c4 * 4);
        }
        // stage B: 32 k x 64 cols
        for (int i = tid; i < 32 * 16; i += 128) {
            int r = i >> 4, c4 = i & 15;
            async_copy16(W + (size_t)(kb + r) * D_ + col0 + c4 * 4,
                         Bs + r * 64 + c4 * 4);
        }
        async_wait();
        __syncthreads();

        for (int kk = 0; kk < 32; kk += 4) {
            const int k2 = kk + 2 * half;
            v2f a[2], b[2];
            for (int r = 0; r < 2; ++r) {
                int m = wr + r * 16 + l15;
                a[r].x = As[m * 32 + k2];
                a[r].y = As[m * 32 + k2 + 1];
            }
            for (int c = 0; c < 2; ++c) {
                int n = wc + c * 16 + l15;
                b[c].x = Bs[k2 * 64 + n];
                b[c].y = Bs[(k2 + 1) * 64 + n];
            }
            for (int r = 0; r < 2; ++r)
                for (int c = 0; c < 2; ++c)
                    acc[r][c] = wmma4(a[r], b[c], acc[r][c]);
        }
        __syncthreads();
    }

    for (int r = 0; r < 2; ++r)
        for (int c = 0; c < 2; ++c) {
            int ncol = col0 + wc + c * 16 + l15;
            for (int e = 0; e < 8; ++e) {
                int mrow = row0 + wr + r * 16 + e + 8 * half;
                O[(size_t)mrow * D_ + ncol] = acc[r][c][e];
            }
        }
}

// ---------------------------------------------------------------------------
// Kernel 2: s[b,t] = (sum_d Q) * (sum_d K).   grid 1024, block 256 (8 rows/blk)
// ---------------------------------------------------------------------------
__global__ void __launch_bounds__(256)
rowsum_kernel(float* __restrict__ ws) {
    const float* Q = ws;
    const float* K = ws + MD_;
    float* sArr = ws + 3 * MD_;

    const int row  = blockIdx.x * 8 + (threadIdx.x >> 5);
    const int lane = threadIdx.x & 31;
    const float* qrow = Q + (size_t)row * D_;
    const float* krow = K + (size_t)row * D_;
    float sq = 0.f, sk = 0.f;
    for (int i = lane; i < D_; i += 32) { sq += qrow[i]; sk += krow[i]; }
    for (int off = 16; off; off >>= 1) {
        sq += __shfl_xor(sq, off, 32);
        sk += __shfl_xor(sk, off, 32);
    }
    if (lane == 0) sArr[row] = sq * sk;
}

// ---------------------------------------------------------------------------
// Kernel 3: chunked retention. grid (B=8, D/64=8), block 256 (8 waves).
// State slice R[512 dk][64 dv] lives in LDS (128 KB) for all 16 chunks.
// V chunk tiles are staged by the Tensor Data Mover (one 64x64 2D D# per
// chunk); Q tiles by async global->LDS b128 copies.
// ---------------------------------------------------------------------------
__global__ void __launch_bounds__(256)
retention_kernel(const float* __restrict__ ws, float* __restrict__ out) {
    __shared__ float Rs[512 * 64];   // 128 KB  [dk][dv]
    __shared__ float Qs[64 * 64];    // [a][dk_sub] shifted queries
    __shared__ float Kt[64 * 64];    // [dk_sub][token]
    __shared__ float Vs[64 * 64];    // [token][dv]
    __shared__ float Ps[64 * 64];    // [a][token] masked scores
    __shared__ float gp[80];         // g^i table

    const int b  = blockIdx.x;
    const int jv = blockIdx.y;       // dv slice index
    const float* Q = ws;
    const float* K = ws + MD_;
    const float* V = ws + 2 * MD_;
    const float* sArr = ws + 3 * MD_;

    const int tid  = threadIdx.x;
    const int lane = tid & 31;
    const int wave = tid >> 5;       // 0..7
    const int half = lane >> 4;
    const int l15  = lane & 15;
    const float g  = GAMMA_F * GAMMA_F;

    if (tid < 80) gp[tid] = powf(g, (float)tid);
    for (int i = tid; i < 512 * 64; i += 256) Rs[i] = 0.f;
    __syncthreads();

    // wave's 16x32 slab of the 64x64 Z / P outputs
    const int zr = (wave & 3) * 16;  // query-row base a
    const int zc = (wave >> 2) * 32; // col base (dv for Z, token for P)
    const v8f vzero = {0.f,0.f,0.f,0.f,0.f,0.f,0.f,0.f};
    const float g64 = gp[64];

    for (int c = 0; c < 16; ++c) {
        const int c0 = c * 64;

        // ---- load V chunk slice: [64 tokens][64 dv] -----------------------
#if USE_TDM
        if (wave == 0) {
            const float* gsrc = V + ((size_t)b * S_ + c0) * D_ + jv * 64;
            unsigned long long ga = (unsigned long long)(size_t)gsrc;
            u32x4 g0;
            g0.x = 1u;                                           // count=1, normal mode
            g0.y = lds_offset(Vs);                               // LDS byte address
            g0.z = (unsigned)(ga & 0xFFFFFFFFull);               // global addr lo
            g0.w = (unsigned)((ga >> 32) & 0x01FFFFFFull) | (2u << 30); // hi | type=2
            i32x8 g1;
            g1[0] = (int)(2u << 16);      // wg_mask=0, data_size=4B
            g1[1] = (int)(512u << 16);    // tensor_dim0 = 512 (lo16)
            g1[2] = (int)(4096u << 16);   // tensor_dim0 hi=0 | tensor_dim1 = 4096 (lo16)
            g1[3] = (int)(64u << 16);     // tensor_dim1 hi=0 | tile_dim0 = 64
            g1[4] = 64;                   // tile_dim1 = 64, tile_dim2 = 0
            g1[5] = 512;                  // tensor_dim0_stride = 512 (lo32)
            g1[6] = 0;                    // stride0 hi | stride1 lo
            g1[7] = 0;
            i32x4 gz = {0, 0, 0, 0};
#if __clang_major__ >= 23
            i32x8 gz8 = {0, 0, 0, 0, 0, 0, 0, 0};
            __builtin_amdgcn_tensor_load_to_lds(g0, g1, gz, gz, gz8, 0);
#else
            __builtin_amdgcn_tensor_load_to_lds(g0, g1, gz, gz, 0);
#endif
            __builtin_amdgcn_s_wait_tensorcnt(0);
        }
        __syncthreads();
#else
        for (int i = tid; i < 64 * 16; i += 256) {
            int t = i >> 4, c4 = i & 15;
            async_copy16(V + ((size_t)b * S_ + c0 + t) * D_ + jv * 64 + c4 * 4,
                         Vs + t * 64 + c4 * 4);
        }
        async_wait();
        __syncthreads();
#endif

        // ---- phase B: Z1 = qs @ R  and  Praw = qs @ K^T, over dk ----------
        v8f accZ[2] = {vzero, vzero};
        v8f accP[2] = {vzero, vzero};
        for (int dkb = 0; dkb < D_; dkb += 64) {
            for (int i = tid; i < 64 * 16; i += 256) {   // shifted queries
                int a = i >> 4, c4 = i & 15;
                int t = c0 + a + 1;
                float* dst = Qs + a * 64 + c4 * 4;
                if (t < S_) {
                    async_copy16(Q + ((size_t)b * S_ + t) * D_ + dkb + c4 * 4, dst);
                } else {
                    float4 z = {0.f, 0.f, 0.f, 0.f};
                    *(float4*)dst = z;
                }
            }
            for (int i = tid; i < 64 * 16; i += 256) {   // K chunk, transposed
                int t = i >> 4, c4 = i & 15;
                float4 v = *(const float4*)(K + ((size_t)b * S_ + c0 + t) * D_ + dkb + c4 * 4);
                Kt[(c4 * 4 + 0) * 64 + t] = v.x;
                Kt[(c4 * 4 + 1) * 64 + t] = v.y;
                Kt[(c4 * 4 + 2) * 64 + t] = v.z;
                Kt[(c4 * 4 + 3) * 64 + t] = v.w;
            }
            async_wait();
            __syncthreads();
            for (int kk = 0; kk < 64; kk += 4) {
                const int k2 = kk + 2 * half;
                v2f aq;
                aq.x = Qs[(zr + l15) * 64 + k2];
                aq.y = Qs[(zr + l15) * 64 + k2 + 1];
                for (int ct = 0; ct < 2; ++ct) {
                    const int n = zc + ct * 16 + l15;
                    v2f bz, bp;
                    bz.x = Rs[(dkb + k2) * 64 + n];       // B for Z: R rows
                    bz.y = Rs[(dkb + k2 + 1) * 64 + n];
                    bp.x = Kt[k2 * 64 + n];               // B for P: K^T
                    bp.y = Kt[(k2 + 1) * 64 + n];
                    accZ[ct] = wmma4(aq, bz, accZ[ct]);
                    accP[ct] = wmma4(aq, bp, accP[ct]);
                }
            }
            __syncthreads();
        }

        // scale inter rows by g^(a+2); mask intra scores with g^(a-b+1)
        for (int ct = 0; ct < 2; ++ct) {
            for (int e = 0; e < 8; ++e) {
                const int a  = zr + e + 8 * half;
                const int bb = zc + ct * 16 + l15;
                accZ[ct][e] *= gp[a + 2];
                float pv = (bb <= a) ? accP[ct][e] * gp[a - bb + 1] : 0.f;
                Ps[a * 64 + bb] = pv;
            }
        }
        __syncthreads();

        // ---- phase C: Z += Pmasked @ V ------------------------------------
        for (int kk = 0; kk < 64; kk += 4) {
            const int k2 = kk + 2 * half;
            v2f ap;
            ap.x = Ps[(zr + l15) * 64 + k2];
            ap.y = Ps[(zr + l15) * 64 + k2 + 1];
            for (int ct = 0; ct < 2; ++ct) {
                const int n = zc + ct * 16 + l15;
                v2f bv;
                bv.x = Vs[k2 * 64 + n];
                bv.y = Vs[(k2 + 1) * 64 + n];
                accZ[ct] = wmma4(ap, bv, accZ[ct]);
            }
        }

        // ---- write output: out = s*V + cross ------------------------------
        for (int ct = 0; ct < 2; ++ct) {
            for (int e = 0; e < 8; ++e) {
                const int a   = zr + e + 8 * half;
                const int dvl = zc + ct * 16 + l15;
                const int t   = c0 + a;
                float o = sArr[(size_t)b * S_ + t] * Vs[a * 64 + dvl] + accZ[ct][e];
                out[((size_t)b * S_ + t) * D_ + jv * 64 + dvl] = o;
            }
        }
        __syncthreads();

        // ---- phase D: R = g^64 * R + sum_t g^(63-t) k_t (x) v_t -----------
        // wave owns dk rows [wave*64, wave*64+64)
        const int dk0 = wave * 64;
        const float* Kg = K + ((size_t)b * S_ + c0) * D_;   // [token][dk]
        for (int ct = 0; ct < 4; ++ct) {
            const int n = ct * 16 + l15;                    // dv col
            v8f acc[4];
            for (int rt = 0; rt < 4; ++rt)
                for (int e = 0; e < 8; ++e)
                    acc[rt][e] = g64 * Rs[(dk0 + rt * 16 + e + 8 * half) * 64 + n];
            for (int kk = 0; kk < 64; kk += 4) {
                const int k2 = kk + 2 * half;               // token index (<=62)
                const float s0 = gp[63 - k2];
                const float s1 = gp[62 - k2];
                v2f bv;
                bv.x = Vs[k2 * 64 + n];
                bv.y = Vs[(k2 + 1) * 64 + n];
                for (int rt = 0; rt < 4; ++rt) {
                    const int m = dk0 + rt * 16 + l15;      // dk row
                    v2f av;
                    av.x = s0 * Kg[(size_t)k2 * D_ + m];
                    av.y = s1 * Kg[(size_t)(k2 + 1) * D_ + m];
                    acc[rt] = wmma4(av, bv, acc[rt]);
                }
            }
            for (int rt = 0; rt < 4; ++rt)
                for (int e = 0; e < 8; ++e)
                    Rs[(dk0 + rt * 16 + e + 8 * half) * 64 + n] = acc[rt][e];
        }
        __syncthreads();
    }
}

// ---------------------------------------------------------------------------
extern "C" void kernel_launch(void* const* d_in, const int* in_sizes, int n_in,
                              void* d_out, int out_size, void* d_ws, size_t ws_size,
                              hipStream_t stream) {
    const float* xq = (const float*)d_in[0];
    const float* xk = (const float*)d_in[1];
    const float* xv = (const float*)d_in[2];
    const float* Wq = (const float*)d_in[3];
    const float* Wk = (const float*)d_in[4];
    const float* Wv = (const float*)d_in[5];
    float* ws  = (float*)d_ws;   // layout: Q | K | V | s   (3*MD_ + B*S floats)
    float* out = (float*)d_out;

    proj_gemm_kernel<<<dim3(S_ * B_ / 64, D_ / 64, 3), 128, 0, stream>>>(
        xq, xk, xv, Wq, Wk, Wv, ws);
    rowsum_kernel<<<dim3(B_ * S_ / 8), 256, 0, stream>>>(ws);
    retention_kernel<<<dim3(B_, D_ / 64), 256, 0, stream>>>(ws, out);
}